// MaxMin_80049600463153
// MI455X (gfx1250) — compile-verified
//
#include <hip/hip_runtime.h>
#include <math.h>

#define B_ROWS   8192
#define IN_DIM   256
#define OUT_DIM  256
#define BT       16      // output rows per block
#define KT       16      // k-chunk staged in LDS (double-buffered)
#define NC       (IN_DIM / KT)   // 16 chunks
#define THREADS  256     // 8 waves (wave32)

typedef __attribute__((ext_vector_type(4))) unsigned int uint32x4_t_;
typedef __attribute__((ext_vector_type(8))) int          int32x8_t_;
typedef __attribute__((ext_vector_type(4))) int          int32x4_t_;

#if __has_builtin(__builtin_amdgcn_tensor_load_to_lds)
#define HAVE_TDM 1
#if __has_include(<hip/amd_detail/amd_gfx1250_TDM.h>)
#define TDM_6ARG 1
#else
#define TDM_6ARG 0
#endif
#else
#define HAVE_TDM 0
#endif

// Single shared struct -> LDS offset 0 for W (TDM D# lds_addr is an absolute
// byte offset into the workgroup's LDS). Total 34 KB static LDS.
struct __align__(16) Smem {
  float W[2][KT][OUT_DIM];   // 2 x 16 KB, byte offsets 0 / 16384
  float M[2][BT][KT];        // 2 x 1 KB,  byte offsets 32768 / 33792
};
#define LDS_W_OFF(b) ((unsigned)(b) * (unsigned)(KT * OUT_DIM * 4))
#define LDS_M_OFF(b) ((unsigned)(2 * KT * OUT_DIM * 4) + (unsigned)(b) * (unsigned)(BT * KT * 4))

// Compiler-level fence: the TDM writes LDS through a *numeric* address inside
// the descriptor, which LLVM's alias analysis cannot see. Escaping the shared
// pointer into a volatile asm with a memory clobber forces LLVM to assume the
// LDS may have been written, so the ds_loads cannot be folded to undef.
// Emits zero machine instructions.
__device__ __forceinline__ void lds_publish(void* p) {
  asm volatile("" : "+v"(p) :: "memory");
}

#if HAVE_TDM
// Issue one 2D tile load Global->LDS via the Tensor Data Mover.
//   tile_d0 = contiguous elements per row, tile_d1 = rows,
//   stride0 = row stride of the global tensor (elements), data = fp32.
__device__ __forceinline__ void tdm_load_tile(unsigned lds_off, const float* gptr,
                                              unsigned tensor_d0, unsigned tensor_d1,
                                              unsigned tile_d0, unsigned tile_d1,
                                              unsigned stride0) {
  unsigned long long ga = (unsigned long long)(const void*)gptr;
  uint32x4_t_ g0;
  g0[0] = 1u;                                          // count = 1 valid descriptor
  g0[1] = lds_off;                                     // lds_addr (bytes)
  g0[2] = (unsigned)(ga & 0xFFFFFFFFu);                // global_addr[31:0]
  g0[3] = (unsigned)((ga >> 32) & 0x01FFFFFFu)         // global_addr[56:32]
        | 0x80000000u;                                 // type = 2 ("image")
  int32x8_t_ g1;
  g1[0] = (int)(2u << 16);                             // data_size=2 -> 4B; wg_mask=0
  g1[1] = (int)((tensor_d0 & 0xFFFFu) << 16);                          // dim0 lo16
  g1[2] = (int)(((tensor_d0 >> 16) & 0xFFFFu) | ((tensor_d1 & 0xFFFFu) << 16));
  g1[3] = (int)(((tensor_d1 >> 16) & 0xFFFFu) | (tile_d0 << 16));      // dim1 hi | tile_d0
  g1[4] = (int)(tile_d1 & 0xFFFFu);                    // tile_d1; tile_d2 = 0
  g1[5] = (int)stride0;                                // tensor_dim0_stride lo32
  g1[6] = 0;
  g1[7] = 0;
  int32x4_t_ z4 = {0, 0, 0, 0};
#if TDM_6ARG
  int32x8_t_ z8 = {0, 0, 0, 0, 0, 0, 0, 0};
  __builtin_amdgcn_tensor_load_to_lds(g0, g1, z4, z4, z8, 0);
#else
  __builtin_amdgcn_tensor_load_to_lds(g0, g1, z4, z4, 0);
#endif
}
#endif

__global__ __launch_bounds__(THREADS)
void tropical_mm_kernel(const float* __restrict__ m,
                        const float* __restrict__ w,
                        float* __restrict__ out) {
  __shared__ Smem s;

  const int tid     = threadIdx.x;
  const int cx      = tid & 63;          // 64 column groups of 4 cols
  const int ty      = tid >> 6;          // 0..3 row group
  const int o0      = cx * 4;
  const int rowBase = blockIdx.x * BT;

  float acc[4][4];
#pragma unroll
  for (int r = 0; r < 4; ++r)
#pragma unroll
    for (int c = 0; c < 4; ++c) acc[r][c] = -INFINITY;

#if HAVE_TDM
  // Prologue: start the DMA pipeline for chunk 0 (W tile + strided M tile).
  if (tid < 32) {
    tdm_load_tile(LDS_W_OFF(0), w,                           OUT_DIM, IN_DIM, OUT_DIM, KT, OUT_DIM);
    tdm_load_tile(LDS_M_OFF(0), m + (size_t)rowBase * IN_DIM, IN_DIM, B_ROWS, KT, BT, IN_DIM);
  }
#endif

#pragma unroll 2                          // makes buf = c&1 a compile-time constant
  for (int c = 0; c < NC; ++c) {
    const int buf = c & 1;
    const int kt  = c * KT;

#if HAVE_TDM
    if (tid < 32) {
      if (c + 1 < NC) {
        // Issue chunk c+1 into the other buffer, then retire chunk c's pair:
        // TDM ops are in-order per wave, so waiting TENSORcnt<=2 completes
        // exactly the two loads for chunk c while c+1's stay in flight.
        tdm_load_tile(LDS_W_OFF(buf ^ 1), w + (size_t)(kt + KT) * OUT_DIM,
                      OUT_DIM, IN_DIM, OUT_DIM, KT, OUT_DIM);
        tdm_load_tile(LDS_M_OFF(buf ^ 1), m + (size_t)rowBase * IN_DIM + kt + KT,
                      IN_DIM, B_ROWS, KT, BT, IN_DIM);
#if __has_builtin(__builtin_amdgcn_s_wait_tensorcnt)
        __builtin_amdgcn_s_wait_tensorcnt(2);
#else
        asm volatile("s_wait_tensorcnt 2" ::: "memory");
#endif
      } else {
#if __has_builtin(__builtin_amdgcn_s_wait_tensorcnt)
        __builtin_amdgcn_s_wait_tensorcnt(0);
#else
        asm volatile("s_wait_tensorcnt 0" ::: "memory");
#endif
      }
      lds_publish(&s);   // TDM has (possibly) written s -> loads must stay
    }
#else
    // Fallback staging: contiguous 16 KB W chunk + 1 KB strided M tile.
    {
      const float* srcW = w + (size_t)kt * OUT_DIM;
      float* dstW = &s.W[buf][0][0];
#pragma unroll
      for (int it = 0; it < (KT * OUT_DIM) / (THREADS * 4); ++it) {
        int idx = (tid + it * THREADS) * 4;
        *(float4*)&dstW[idx] = *(const float4*)&srcW[idx];
      }
      if (tid < (BT * KT) / 4) {
        int r  = tid / (KT / 4);
        int kq = (tid % (KT / 4)) * 4;
        *(float4*)&s.M[buf][r][kq] =
            *(const float4*)&m[(size_t)(rowBase + r) * IN_DIM + kt + kq];
      }
    }
#endif
    __syncthreads();   // chunk c visible in s.W[buf]/s.M[buf]
    lds_publish(&s);   // all threads: LDS contents defined by an unseen writer

    // ---- VALU core: acc[r][c] = max(acc, min(m, w)) over this k-chunk ----
#pragma unroll
    for (int k4 = 0; k4 < KT; k4 += 4) {
      float mvals[4][4];
#pragma unroll
      for (int r = 0; r < 4; ++r) {
        float4 t = *(const float4*)&s.M[buf][ty * 4 + r][k4];  // broadcast read
        mvals[r][0] = t.x; mvals[r][1] = t.y; mvals[r][2] = t.z; mvals[r][3] = t.w;
      }
#pragma unroll
      for (int kk = 0; kk < 4; ++kk) {
        float4 wv = *(const float4*)&s.W[buf][k4 + kk][o0];    // ds_load_b128
#pragma unroll
        for (int r = 0; r < 4; ++r) {
          float mv = mvals[r][kk];
          acc[r][0] = fmaxf(acc[r][0], fminf(mv, wv.x));       // -> v_maxmin_num_f32
          acc[r][1] = fmaxf(acc[r][1], fminf(mv, wv.y));
          acc[r][2] = fmaxf(acc[r][2], fminf(mv, wv.z));
          acc[r][3] = fmaxf(acc[r][3], fminf(mv, wv.w));
        }
      }
    }
    __syncthreads();   // all reads of buf done before it is refilled (iter c+2)
  }

  // ---- store the 4x4 register tile ----
#pragma unroll
  for (int r = 0; r < 4; ++r) {
    float4 o;
    o.x = acc[r][0]; o.y = acc[r][1]; o.z = acc[r][2]; o.w = acc[r][3];
    *(float4*)&out[(size_t)(rowBase + ty * 4 + r) * OUT_DIM + o0] = o;
  }
}

extern "C" void kernel_launch(void* const* d_in, const int* in_sizes, int n_in,
                              void* d_out, int out_size, void* d_ws, size_t ws_size,
                              hipStream_t stream) {
  (void)in_sizes; (void)n_in; (void)out_size; (void)d_ws; (void)ws_size;
  const float* m = (const float*)d_in[0];      // [8192, 256] fp32
  const float* w = (const float*)d_in[1];      // [256, 256] fp32
  float* out     = (float*)d_out;              // [8192, 256] fp32

  dim3 grid(B_ROWS / BT);                      // 512 blocks
  dim3 block(THREADS);                         // 8 wave32 waves
  tropical_mm_kernel<<<grid, block, 0, stream>>>(m, w, out);
}